// roi_pooling_20444044329268
// MI455X (gfx1250) — compile-verified
//
#include <hip/hip_runtime.h>
#include <math.h>

// ROI max-pool, MI455X/gfx1250.
// Grid (P, P, B*N): hardware decodes (px, py, roi) -> no div/mod index math.
// One block (128 threads = 4 wave32) per (roi, bin) cell; thread t owns
// channels [4t, 4t+4), so one block's lanes cover a full 2KB pixel channel
// line with coalesced global_load_b128. Max-reduce in VGPRs (v_max_num_f32),
// next-row global_prefetch_b8, non-temporal b128 stores for the
// write-once output so L2 stays dedicated to the 5.9MB feature map.

namespace {
constexpr int kB = 2, kN = 64, kH = 38, kW = 38, kC = 512, kP = 7;
constexpr int kCells   = kP * kP;   // 49
constexpr int kThreads = kC / 4;    // 128 threads, one float4 of channels each
}

typedef __attribute__((ext_vector_type(4))) float v4f;
typedef __attribute__((ext_vector_type(4))) int   v4i;

__global__ __launch_bounds__(kThreads)
void roi_maxpool_gfx1250_kernel(const float* __restrict__ fmap,
                                const int*  __restrict__ rois,
                                float* __restrict__ out)
{
    const int px  = blockIdx.x;           // 0..6
    const int py  = blockIdx.y;           // 0..6
    const int roi = blockIdx.z;           // 0..B*N-1
    const int b   = roi >> 6;             // kN == 64

    // ROI quad (y, x, h, w) — block-uniform, 16B aligned -> scalar s_load_b128.
    const v4i r  = ((const v4i*)rois)[roi];
    const int y0 = r.x, x0 = r.y, h = r.z, w = r.w;

    // Exact bin partition [floor(p*h/P), floor((p+1)*h/P)); h,w >= P so every
    // bin is non-empty (matches the reference's ceil-div bin formula).
    // Unsigned div-by-7 -> single mulhi+shift per bound.
    const int ys = y0 + (int)((unsigned)(py * h) / 7u);
    const int ye = y0 + (int)((unsigned)((py + 1) * h) / 7u);
    const int xs = x0 + (int)((unsigned)(px * w) / 7u);
    const int xe = x0 + (int)((unsigned)((px + 1) * w) / 7u);

    const int    c4    = threadIdx.x;     // channel-quad id: 0..127
    const float* bbase = fmap + (size_t)b * (kH * kW * kC) + (size_t)c4 * 4;

    v4f acc = (v4f){-INFINITY, -INFINITY, -INFINITY, -INFINITY};

    for (int y = ys; y < ye; ++y) {
        const float* rowp = bbase + (size_t)y * (kW * kC);

        // Pull the next ROI row's first pixel toward the caches while this
        // row's short (<=6 pixel) loop drains -> global_prefetch_b8.
        if (y + 1 < ye) {
            __builtin_prefetch(rowp + (size_t)(kW * kC) + (size_t)xs * kC, 0, 2);
        }

        int x = xs;
        // Unroll x2: two independent b128 loads in flight per iteration.
        for (; x + 1 < xe; x += 2) {
            v4f v0 = *(const v4f*)(rowp + (size_t)x * kC);
            v4f v1 = *(const v4f*)(rowp + (size_t)(x + 1) * kC);
            acc.x = fmaxf(acc.x, fmaxf(v0.x, v1.x));
            acc.y = fmaxf(acc.y, fmaxf(v0.y, v1.y));
            acc.z = fmaxf(acc.z, fmaxf(v0.z, v1.z));
            acc.w = fmaxf(acc.w, fmaxf(v0.w, v1.w));
        }
        if (x < xe) {
            v4f v0 = *(const v4f*)(rowp + (size_t)x * kC);
            acc.x = fmaxf(acc.x, v0.x);
            acc.y = fmaxf(acc.y, v0.y);
            acc.z = fmaxf(acc.z, v0.z);
            acc.w = fmaxf(acc.w, v0.w);
        }
    }

    // out[roi][py][px][C], flat cell index without any division.
    const int flat = roi * kCells + py * kP + px;
    v4f* op = (v4f*)(out + (size_t)flat * kC) + c4;
    __builtin_nontemporal_store(acc, op);
}

extern "C" void kernel_launch(void* const* d_in, const int* in_sizes, int n_in,
                              void* d_out, int out_size, void* d_ws, size_t ws_size,
                              hipStream_t stream) {
    (void)in_sizes; (void)n_in; (void)d_ws; (void)ws_size; (void)out_size;
    const float* fmap = (const float*)d_in[0];   // [B,H,W,C] f32
    const int*   rois = (const int*)d_in[1];     // [B,N,4]   i32
    float*       out  = (float*)d_out;           // [B,N,P,P,C] f32

    roi_maxpool_gfx1250_kernel<<<dim3(kP, kP, kB * kN), dim3(kThreads), 0, stream>>>(
        fmap, rois, out);
}